// OneDRAC_Pose_2113123909969
// MI455X (gfx1250) — compile-verified
//
#include <hip/hip_runtime.h>
#include <hip/hip_bf16.h>
#include <math.h>
#include <stdint.h>

typedef __attribute__((ext_vector_type(2))) float v2f;
typedef __attribute__((ext_vector_type(8))) float v8f;

// ---------------------------------------------------------------- activation
__device__ __forceinline__ float actf(float v, int act) {
  switch (act) {
    case 1: return v > 0.f ? v : 0.f;                                   // relu
    case 2: return 0.5f * v * (1.f + erff(v * 0.70710678118654752f));   // gelu exact
    case 3: return v > 0.f ? v : 0.01f * v;                             // leaky
    default: return v;
  }
}

// ---------------------------------------------------------------- CDNA5 async global->LDS copies
__device__ __forceinline__ void async_g2l_b128(const void* lds, const float* g) {
  unsigned la = (unsigned)(uintptr_t)lds;
  unsigned long long ga = (unsigned long long)(uintptr_t)g;
  asm volatile("global_load_async_to_lds_b128 %0, %1, off" :: "v"(la), "v"(ga) : "memory");
}
__device__ __forceinline__ void async_g2l_b32(const void* lds, const float* g) {
  unsigned la = (unsigned)(uintptr_t)lds;
  unsigned long long ga = (unsigned long long)(uintptr_t)g;
  asm volatile("global_load_async_to_lds_b32 %0, %1, off" :: "v"(la), "v"(ga) : "memory");
}
__device__ __forceinline__ void wait_async0() {
  asm volatile("s_wait_asynccnt 0x0" ::: "memory");
}

// ---------------------------------------------------------------- WMMA GEMM
// C[b1,b2] = act(alpha * A (MxK) * B (KxN or NxK) + bias) + Res
// f32 WMMA 16x16x4; block tile 128x64; 8 wave32 waves; wave tile 16x64 (4 acc).
// LDS tiles [row][k], 20-float padded stride (16B-aligned rows, conflict-free
// b64 fragment reads). Double-buffered: tile i+1 is staged through the gfx1250
// async global->LDS engine (ASYNCcnt) while tile i runs on the WMMA pipe.
__global__ __launch_bounds__(256) void k_gemm(
    const float* __restrict__ A, const float* __restrict__ B,
    const float* __restrict__ bias, const float* __restrict__ Res,
    float* __restrict__ C,
    int M, int N, int K, int lda, int ldb, int ldc,
    long sA1, long sA2, long sB1, long sB2, long sC1, long sC2,
    int b2n, int transB, float alpha, int act)
{
  __shared__ float sA[2][128][20];   // [buf][m][k]
  __shared__ float sB[2][64][20];    // [buf][n][k]

  int bz = blockIdx.z;
  int b1 = bz / b2n, b2 = bz - b1 * b2n;
  A += (long)b1 * sA1 + (long)b2 * sA2;
  B += (long)b1 * sB1 + (long)b2 * sB2;
  C += (long)b1 * sC1 + (long)b2 * sC2;
  const float* Rp = Res ? (Res + (long)b1 * sC1 + (long)b2 * sC2) : nullptr;

  int m0 = blockIdx.y * 128;
  int n0 = blockIdx.x * 64;
  int t = threadIdx.x;
  int lane = t & 31;
  int wave = t >> 5;            // 0..7 -> 16-row strip
  int lh = lane >> 4;           // 0/1
  int l15 = lane & 15;
  int am = wave * 16 + l15;

  v8f acc[4];
  #pragma unroll
  for (int i = 0; i < 4; ++i) acc[i] = (v8f){0.f,0.f,0.f,0.f,0.f,0.f,0.f,0.f};

  // loader index precompute (256 threads)
  int a_m = t >> 1, a_kq = (t & 1) * 8;       // A: 128 rows x 16k, 8 elems/thread
  int bt_n = t >> 2, bt_kq = (t & 3) * 4;     // B (NxK): 64 rows x 16k, 4 elems/thread
  int bn_k = t >> 4, bn_nq = (t & 15) * 4;    // B (KxN): 16 rows x 64n, 4 elems/thread

  bool intA = (m0 + 128 <= M) && ((lda & 3) == 0);
  bool intB = (n0 + 64 <= N) && (!transB || ((ldb & 3) == 0));

  // stage one 16-deep K-tile into LDS buffer bi
  auto stage = [&](int kb, int bi) {
    bool kin = (kb + 16 <= K);
    // ---- A
    if (intA && kin) {
      const float* ga = A + (long)(m0 + a_m) * lda + kb + a_kq;
      if (kb + 48 <= K) __builtin_prefetch(ga + 32, 0, 3);
      async_g2l_b128(&sA[bi][a_m][a_kq], ga);
      async_g2l_b128(&sA[bi][a_m][a_kq + 4], ga + 4);
    } else {
      int grow = m0 + a_m;
      #pragma unroll
      for (int j = 0; j < 8; ++j) {
        int kg = kb + a_kq + j;
        float v = 0.f;
        if (grow < M && kg < K) v = A[(long)grow * lda + kg];
        sA[bi][a_m][a_kq + j] = v;
      }
    }
    // ---- B
    if (intB && kin) {
      if (!transB) {
        const float* gb = B + (long)(kb + bn_k) * ldb + n0 + bn_nq;
        #pragma unroll
        for (int j = 0; j < 4; ++j)
          async_g2l_b32(&sB[bi][bn_nq + j][bn_k], gb + j);
      } else {
        const float* gb = B + (long)(n0 + bt_n) * ldb + kb + bt_kq;
        async_g2l_b128(&sB[bi][bt_n][bt_kq], gb);
      }
    } else if (!transB) {
      int kg = kb + bn_k;
      #pragma unroll
      for (int j = 0; j < 4; ++j) {
        int ng = n0 + bn_nq + j;
        float v = 0.f;
        if (kg < K && ng < N) v = B[(long)kg * ldb + ng];
        sB[bi][bn_nq + j][bn_k] = v;
      }
    } else {
      int ng = n0 + bt_n;
      #pragma unroll
      for (int j = 0; j < 4; ++j) {
        int kg = kb + bt_kq + j;
        float v = 0.f;
        if (ng < N && kg < K) v = B[(long)ng * ldb + kg];
        sB[bi][bt_n][bt_kq + j] = v;
      }
    }
  };

  int nt = (K + 15) / 16;
  stage(0, 0);
  for (int i = 0; i < nt; ++i) {
    int cur = i & 1;
    wait_async0();          // tile i's async copies complete (issued one phase ago)
    __syncthreads();        // tile i visible to all waves
    if (i + 1 < nt) stage((i + 1) * 16, cur ^ 1);   // overlap with compute below

    #pragma unroll
    for (int kk = 0; kk < 16; kk += 4) {
      int k2 = kk + lh * 2;
      v2f a = *(const v2f*)&sA[cur][am][k2];
      #pragma unroll
      for (int j = 0; j < 4; ++j) {
        v2f bb = *(const v2f*)&sB[cur][j * 16 + l15][k2];
        acc[j] = __builtin_amdgcn_wmma_f32_16x16x4_f32(false, a, false, bb, (short)0, acc[j], false, false);
      }
    }
    __syncthreads();        // all waves done reading buffer `cur`
  }

  // ---- epilogue
  #pragma unroll
  for (int i = 0; i < 4; ++i) {
    int col = n0 + i * 16 + l15;
    if (col < N) {
      #pragma unroll
      for (int r = 0; r < 8; ++r) {
        int row = m0 + wave * 16 + r + lh * 8;
        if (row < M) {
          float v = alpha * acc[i][r];
          if (bias) v += bias[col];
          v = actf(v, act);
          if (Rp) v += Rp[(long)row * ldc + col];
          C[(long)row * ldc + col] = v;
        }
      }
    }
  }
}

// ---------------------------------------------------------------- softmax (per row)
__global__ __launch_bounds__(128) void k_softmax(float* __restrict__ S, int N) {
  long row = blockIdx.x;
  float* r = S + row * (long)N;
  __shared__ float red[128];
  int t = threadIdx.x;
  float m = -3.4e38f;
  for (int i = t; i < N; i += 128) m = fmaxf(m, r[i]);
  red[t] = m; __syncthreads();
  for (int o = 64; o > 0; o >>= 1) { if (t < o) red[t] = fmaxf(red[t], red[t + o]); __syncthreads(); }
  m = red[0]; __syncthreads();
  float s = 0.f;
  for (int i = t; i < N; i += 128) { float e = expf(r[i] - m); r[i] = e; s += e; }
  red[t] = s; __syncthreads();
  for (int o = 64; o > 0; o >>= 1) { if (t < o) red[t] += red[t + o]; __syncthreads(); }
  float inv = 1.f / red[0];
  for (int i = t; i < N; i += 128) r[i] *= inv;
}

// ---------------------------------------------------------------- (X [+R]) -> LayerNorm
__global__ __launch_bounds__(128) void k_add_ln(
    const float* __restrict__ X, const float* __restrict__ R,
    const float* __restrict__ g, const float* __restrict__ b,
    float* __restrict__ O, int D)
{
  long row = blockIdx.x;
  const float* x = X + row * (long)D;
  const float* rr = R ? R + row * (long)D : nullptr;
  float* o = O + row * (long)D;
  __shared__ float red[128];
  int t = threadIdx.x;
  float vals[4];
  float s = 0.f;
  int c = 0;
  for (int i = t; i < D; i += 128) { float v = x[i] + (rr ? rr[i] : 0.f); vals[c++] = v; s += v; }
  red[t] = s; __syncthreads();
  for (int o2 = 64; o2 > 0; o2 >>= 1) { if (t < o2) red[t] += red[t + o2]; __syncthreads(); }
  float mean = red[0] / (float)D;
  __syncthreads();
  float s2 = 0.f; c = 0;
  for (int i = t; i < D; i += 128) { float dv = vals[c++] - mean; s2 += dv * dv; }
  red[t] = s2; __syncthreads();
  for (int o2 = 64; o2 > 0; o2 >>= 1) { if (t < o2) red[t] += red[t + o2]; __syncthreads(); }
  float inv = rsqrtf(red[0] / (float)D + 1e-5f);
  c = 0;
  for (int i = t; i < D; i += 128) o[i] = (vals[c++] - mean) * inv * g[i] + b[i];
}

// ---------------------------------------------------------------- spatial: x + pe3
__global__ void k_prep_spatial(const float* __restrict__ X, float* __restrict__ O, int total) {
  int i = blockIdx.x * blockDim.x + threadIdx.x;
  if (i >= total) return;
  int row = i / 51; int cc = i % 51; int d = cc % 3;
  float pos = (float)row;
  float pe = (d == 0) ? sinf(pos) : (d == 1 ? cosf(pos) : 0.f);
  O[i] = X[i] + pe;
}

// ---------------------------------------------------------------- spatial MHA (d=3,h=3,dk=1), out = x + attn
__global__ void k_spatial_mha(const float* __restrict__ X,
    const float* wq, const float* bq, const float* wk, const float* bk,
    const float* wv, const float* bv, const float* wo, const float* bo,
    float* __restrict__ O)
{
  int blk = blockIdx.x;                 // 4096 tokens
  const float* x = X + (long)blk * 51;
  __shared__ float xs[17][3], qs[17][3], ks[17][3], vs[17][3];
  int t = threadIdx.x;                  // 32
  if (t < 17) { for (int d = 0; d < 3; ++d) xs[t][d] = x[t * 3 + d]; }
  __syncthreads();
  if (t < 17) {
    for (int cdim = 0; cdim < 3; ++cdim) {
      float q = bq[cdim], k = bk[cdim], v = bv[cdim];
      for (int j = 0; j < 3; ++j) {
        float xv = xs[t][j];
        q += xv * wq[j * 3 + cdim]; k += xv * wk[j * 3 + cdim]; v += xv * wv[j * 3 + cdim];
      }
      qs[t][cdim] = q; ks[t][cdim] = k; vs[t][cdim] = v;
    }
  }
  __syncthreads();
  if (t < 17) {
    float oh[3];
    for (int h = 0; h < 3; ++h) {
      float mx = -3.4e38f, sc[17];
      for (int j = 0; j < 17; ++j) { sc[j] = qs[t][h] * ks[j][h]; mx = fmaxf(mx, sc[j]); }
      float s = 0.f;
      for (int j = 0; j < 17; ++j) { float e = expf(sc[j] - mx); sc[j] = e; s += e; }
      float a = 0.f;
      for (int j = 0; j < 17; ++j) a += sc[j] * vs[j][h];
      oh[h] = a / s;
    }
    for (int d = 0; d < 3; ++d) {
      float r = bo[d];
      for (int h = 0; h < 3; ++h) r += oh[h] * wo[h * 3 + d];
      O[(long)blk * 51 + t * 3 + d] = xs[t][d] + r;
    }
  }
}

// ---------------------------------------------------------------- spatial FFN fused: out = x + relu(x@w1+b1)@w2+b2
__global__ __launch_bounds__(128) void k_spatial_ffn(const float* __restrict__ X,
    const float* __restrict__ w1, const float* __restrict__ b1,
    const float* __restrict__ w2, const float* __restrict__ b2,
    float* __restrict__ O)
{
  long row = blockIdx.x;            // 69632 rows, D=3, FF=512
  const float* x = X + row * 3;
  float x0 = x[0], x1 = x[1], x2 = x[2];
  float p0 = 0.f, p1 = 0.f, p2 = 0.f;
  int t = threadIdx.x;
  for (int j = t; j < 512; j += 128) {
    float h = b1[j] + x0 * w1[j] + x1 * w1[512 + j] + x2 * w1[1024 + j];
    h = fmaxf(h, 0.f);
    p0 += h * w2[j * 3]; p1 += h * w2[j * 3 + 1]; p2 += h * w2[j * 3 + 2];
  }
  __shared__ float r0[128], r1[128], r2[128];
  r0[t] = p0; r1[t] = p1; r2[t] = p2; __syncthreads();
  for (int o = 64; o > 0; o >>= 1) {
    if (t < o) { r0[t] += r0[t + o]; r1[t] += r1[t + o]; r2[t] += r2[t + o]; }
    __syncthreads();
  }
  if (t == 0) {
    O[row * 3 + 0] = x0 + b2[0] + r0[0];
    O[row * 3 + 1] = x1 + b2[1] + r1[0];
    O[row * 3 + 2] = x2 + b2[2] + r2[0];
  }
}

// ---------------------------------------------------------------- concat [x | spatial] -> (4096,102)
__global__ void k_concat(const float* __restrict__ X, const float* __restrict__ SP,
                         float* __restrict__ O, int total) {
  int i = blockIdx.x * blockDim.x + threadIdx.x;
  if (i >= total) return;
  int row = i / 102, cc = i % 102;
  O[i] = (cc < 51) ? X[(long)row * 51 + cc] : SP[(long)row * 51 + (cc - 51)];
}

// ---------------------------------------------------------------- DFT tables (packed real|imag, ortho)
__global__ void k_build_dft(float* __restrict__ Wf, float* __restrict__ Wi) {
  int i = blockIdx.x * blockDim.x + threadIdx.x;
  const double inv = 0.04419417382415922;     // 1/sqrt(512)
  const double w0 = 6.283185307179586 / 512.0;
  if (i < 512 * 514) {
    int d = i / 514, cc = i % 514;
    int f = (cc < 257) ? cc : (cc - 257);
    double ang = w0 * (double)d * (double)f;
    Wf[i] = (float)(((cc < 257) ? cos(ang) : -sin(ang)) * inv);
    // Wi has the same element count (514*512)
    int fr = i / 512, n = i % 512;
    int fi = (fr < 257) ? fr : (fr - 257);
    double wgt = (fi == 0 || fi == 256) ? 1.0 : 2.0;
    double ang2 = w0 * (double)fi * (double)n;
    Wi[i] = (float)(((fr < 257) ? cos(ang2) : -sin(ang2)) * wgt * inv);
  }
}

// ---------------------------------------------------------------- out = in + pe512[b,:] (broadcast over T)
__global__ void k_add_pe512(const float* __restrict__ In, float* __restrict__ O, int total) {
  int i = blockIdx.x * blockDim.x + threadIdx.x;
  if (i >= total) return;
  int d = i & 511;
  int row = i >> 9;
  int b = row >> 7;                 // batch index (row / 128)
  int ii = d >> 1;
  float div = expf(-0.017988946039f * (float)(2 * ii));   // ln(1e4)/512
  float a = (float)b * div;
  float pe = (d & 1) ? cosf(a) : sinf(a);
  O[i] = In[i] + pe;
}

// ---------------------------------------------------------------- y = x1[:,32:96,:]
__global__ void k_slice_y(const float* __restrict__ X, float* __restrict__ Y, int total) {
  int i = blockIdx.x * blockDim.x + threadIdx.x;
  if (i >= total) return;
  int b = i / (64 * 512);
  int rem = i - b * 64 * 512;
  int r = rem / 512, d = rem - r * 512;
  Y[i] = X[((long)(b * 128 + 32 + r)) * 512 + d];
}

// ================================================================ host side
struct EncP {
  const float *wq, *bq, *wk, *bk, *wv, *bv, *wo, *bo;
  const float *w1, *b1, *w2, *b2, *l1g, *l1b, *l2g, *l2b;
};

static inline void gemmL(hipStream_t st, const float* A, const float* B, const float* bias,
                         const float* Res, float* C, int M, int N, int K,
                         int lda, int ldb, int ldc,
                         long sA1, long sA2, long sB1, long sB2, long sC1, long sC2,
                         int b1n, int b2n, int transB, float alpha, int act) {
  dim3 g((N + 63) / 64, (M + 127) / 128, b1n * b2n);
  k_gemm<<<g, 256, 0, st>>>(A, B, bias, Res, C, M, N, K, lda, ldb, ldc,
                            sA1, sA2, sB1, sB2, sC1, sC2, b2n, transB, alpha, act);
}
static inline void gemmP(hipStream_t st, const float* A, const float* B, const float* bias,
                         const float* Res, float* C, int M, int N, int K, float alpha, int act) {
  gemmL(st, A, B, bias, Res, C, M, N, K, K, N, N, 0, 0, 0, 0, 0, 0, 1, 1, 0, alpha, act);
}

extern "C" void kernel_launch(void* const* d_in, const int* in_sizes, int n_in,
                              void* d_out, int out_size, void* d_ws, size_t ws_size,
                              hipStream_t stream) {
  (void)in_sizes; (void)n_in; (void)out_size; (void)ws_size;
  hipStream_t st = stream;

  // ---- parse params in setup_inputs() insertion order
  int gi = 0;
  auto nx = [&]() { return (const float*)d_in[gi++]; };
  const float* X = nx();
  const float* fc0_w = nx(); const float* fc0_b = nx();
  auto fillE = [&](EncP& p) {
    p.wq = nx(); p.bq = nx(); p.wk = nx(); p.bk = nx();
    p.wv = nx(); p.bv = nx(); p.wo = nx(); p.bo = nx();
    p.w1 = nx(); p.b1 = nx(); p.w2 = nx(); p.b2 = nx();
    p.l1g = nx(); p.l1b = nx(); p.l2g = nx(); p.l2b = nx();
  };
  auto fillA = [&](EncP& p) {
    p.wq = nx(); p.bq = nx(); p.wk = nx(); p.bk = nx();
    p.wv = nx(); p.bv = nx(); p.wo = nx(); p.bo = nx();
  };
  EncP sp[2]; for (int i = 0; i < 2; ++i) fillE(sp[i]);
  EncP fo[6]; for (int i = 0; i < 6; ++i) fillE(fo[i]);
  EncP cr[3]; for (int i = 0; i < 3; ++i) fillE(cr[i]);
  EncP wnp[3]; for (int i = 0; i < 3; ++i) fillA(wnp[i]);
  const float* ae_lng = nx(); const float* ae_lnb = nx();
  const float *ew0 = nx(), *eb0 = nx(), *ew1 = nx(), *eb1 = nx(), *ew2 = nx(), *eb2 = nx();
  const float *dw0 = nx(), *db0 = nx(), *dw1 = nx(), *db1 = nx(),
              *dw2 = nx(), *db2 = nx(), *dw3 = nx(), *db3 = nx();
  EncP te; fillE(te);

  // ---- workspace arena
  size_t off = 0;
  auto alloc = [&](size_t n) { float* p = (float*)d_ws + off; off += (n + 63) & ~(size_t)63; return p; };
  const int LQ = 516;                     // padded row stride for packed-DFT buffers (16B aligned)
  float* spXR  = alloc(4096 * 51);
  float* spTMP = alloc(4096 * 51);
  float* featIn = alloc(4096 * 102);
  float* featB = alloc(4096 * 512);
  float* xA = alloc(4096 * 512);
  float* xB = alloc(4096 * 512);
  float* tmpA = alloc(4096 * 512);
  float* bufQ = alloc(4096 * 512);
  float* bufK = alloc(4096 * 512);
  float* bufV = alloc(4096 * 512);
  float* qri = alloc((size_t)4096 * LQ);
  float* kri = alloc((size_t)4096 * LQ);
  float* vri = alloc((size_t)4096 * LQ);
  float* ori = alloc((size_t)4096 * LQ);
  float* scoreB = alloc((size_t)32 * 8 * 128 * 128);
  float* ffnB = alloc((size_t)4096 * 2048);
  float* yB = alloc((size_t)32 * 64 * 512);
  float* Wfwd = alloc(512 * 514);
  float* Winv = alloc(514 * 512);

  const float ISQ512 = 0.04419417382415922f;   // 1/sqrt(512)
  const float ISQ128 = 0.08838834764831845f;   // 1/sqrt(128)

  // ---- spatial stack (d=3)
  k_prep_spatial<<<(4096 * 51 + 255) / 256, 256, 0, st>>>(X, spXR, 4096 * 51);
  for (int l = 0; l < 2; ++l) {
    const EncP& p = sp[l];
    k_spatial_mha<<<4096, 32, 0, st>>>(spXR, p.wq, p.bq, p.wk, p.bk, p.wv, p.bv, p.wo, p.bo, spTMP);
    k_add_ln<<<69632, 128, 0, st>>>(spTMP, nullptr, p.l1g, p.l1b, spXR, 3);
    k_spatial_ffn<<<69632, 128, 0, st>>>(spXR, p.w1, p.b1, p.w2, p.b2, spTMP);
    k_add_ln<<<69632, 128, 0, st>>>(spTMP, nullptr, p.l2g, p.l2b, spXR, 3);
  }
  k_concat<<<(4096 * 102 + 255) / 256, 256, 0, st>>>(X, spXR, featIn, 4096 * 102);
  gemmP(st, featIn, fc0_w, fc0_b, nullptr, featB, 4096, 512, 102, 1.f, 0);

  // ---- Fourier layers (rfft/irfft as packed DFT GEMMs, padded stride 516)
  k_build_dft<<<(512 * 514 + 255) / 256, 256, 0, st>>>(Wfwd, Winv);
  for (int l = 0; l < 6; ++l) {
    const EncP& p = fo[l];
    gemmP(st, featB, p.wq, p.bq, nullptr, bufQ, 4096, 512, 512, 1.f, 0);
    gemmP(st, featB, p.wk, p.bk, nullptr, bufK, 4096, 512, 512, 1.f, 0);
    gemmP(st, featB, p.wv, p.bv, nullptr, bufV, 4096, 512, 512, 1.f, 0);
    gemmL(st, bufQ, Wfwd, nullptr, nullptr, qri, 4096, 514, 512, 512, 514, LQ,
          0, 0, 0, 0, 0, 0, 1, 1, 0, 1.f, 0);
    gemmL(st, bufK, Wfwd, nullptr, nullptr, kri, 4096, 514, 512, 512, 514, LQ,
          0, 0, 0, 0, 0, 0, 1, 1, 0, 1.f, 0);
    gemmL(st, bufV, Wfwd, nullptr, nullptr, vri, 4096, 514, 512, 512, 514, LQ,
          0, 0, 0, 0, 0, 0, 1, 1, 0, 1.f, 0);
    gemmL(st, qri, kri, nullptr, nullptr, scoreB, 128, 128, 514,
          LQ, LQ, 128, 128L * LQ, 0, 128L * LQ, 0, 128L * 128, 0,
          32, 1, 1, ISQ512, 0);
    k_softmax<<<4096, 128, 0, st>>>(scoreB, 128);
    gemmL(st, scoreB, vri, nullptr, nullptr, ori, 128, 514, 128,
          128, LQ, LQ, 128L * 128, 0, 128L * LQ, 0, 128L * LQ, 0,
          32, 1, 0, 1.f, 0);
    gemmL(st, ori, Winv, nullptr, nullptr, tmpA, 4096, 512, 514, LQ, 512, 512,
          0, 0, 0, 0, 0, 0, 1, 1, 0, 1.f, 0);
    gemmP(st, tmpA, p.wo, p.bo, featB, bufQ, 4096, 512, 512, 1.f, 0);
    k_add_ln<<<4096, 128, 0, st>>>(bufQ, nullptr, p.l1g, p.l1b, featB, 512);
    gemmP(st, featB, p.w1, p.b1, nullptr, ffnB, 4096, 2048, 512, 1.f, 1);
    gemmP(st, ffnB, p.w2, p.b2, featB, bufQ, 4096, 512, 2048, 1.f, 0);
    k_add_ln<<<4096, 128, 0, st>>>(bufQ, nullptr, p.l2g, p.l2b, featB, 512);
  }

  // ---- x1 = feat + pe512 ; y = x1[:,32:96,:]
  k_add_pe512<<<(4096 * 512 + 255) / 256, 256, 0, st>>>(featB, xA, 4096 * 512);
  k_slice_y<<<(32 * 64 * 512 + 255) / 256, 256, 0, st>>>(xA, yB, 32 * 64 * 512);

  float* xc = xA; float* xo = xB;
  // ---- cross blocks (h=4, dk=128, Sy=64, pre-LN, exact gelu)
  for (int l = 0; l < 3; ++l) {
    const EncP& p = cr[l];
    k_add_ln<<<4096, 128, 0, st>>>(xc, nullptr, p.l1g, p.l1b, tmpA, 512);
    gemmP(st, tmpA, p.wq, p.bq, nullptr, bufQ, 4096, 512, 512, 1.f, 0);
    gemmP(st, yB, p.wk, p.bk, nullptr, bufK, 2048, 512, 512, 1.f, 0);
    gemmP(st, yB, p.wv, p.bv, nullptr, bufV, 2048, 512, 512, 1.f, 0);
    gemmL(st, bufQ, bufK, nullptr, nullptr, scoreB, 128, 64, 128,
          512, 512, 64, 128L * 512, 128, 64L * 512, 128, 4L * 128 * 64, 128L * 64,
          32, 4, 1, ISQ128, 0);
    k_softmax<<<32 * 4 * 128, 128, 0, st>>>(scoreB, 64);
    gemmL(st, scoreB, bufV, nullptr, nullptr, tmpA, 128, 128, 64,
          64, 512, 512, 4L * 128 * 64, 128L * 64, 64L * 512, 128, 128L * 512, 128,
          32, 4, 0, 1.f, 0);
    gemmP(st, tmpA, p.wo, p.bo, xc, xo, 4096, 512, 512, 1.f, 0);
    { float* s = xc; xc = xo; xo = s; }
    k_add_ln<<<4096, 128, 0, st>>>(xc, nullptr, p.l2g, p.l2b, tmpA, 512);
    gemmP(st, tmpA, p.w1, p.b1, nullptr, ffnB, 4096, 2048, 512, 1.f, 2);
    gemmP(st, ffnB, p.w2, p.b2, xc, xo, 4096, 512, 2048, 1.f, 0);
    { float* s = xc; xc = xo; xo = s; }
  }

  // ---- window attention (w=4, h=4, dk=128)
  for (int l = 0; l < 3; ++l) {
    const EncP& p = wnp[l];
    gemmP(st, xc, p.wq, p.bq, nullptr, bufQ, 4096, 512, 512, 1.f, 0);
    gemmP(st, xc, p.wk, p.bk, nullptr, bufK, 4096, 512, 512, 1.f, 0);
    gemmP(st, xc, p.wv, p.bv, nullptr, bufV, 4096, 512, 512, 1.f, 0);
    gemmL(st, bufQ, bufK, nullptr, nullptr, scoreB, 4, 4, 128,
          512, 512, 4, 4L * 512, 128, 4L * 512, 128, 4L * 16, 16,
          1024, 4, 1, ISQ128, 0);
    k_softmax<<<1024 * 4 * 4, 128, 0, st>>>(scoreB, 4);
    gemmL(st, scoreB, bufV, nullptr, nullptr, tmpA, 4, 128, 4,
          4, 512, 512, 4L * 16, 16, 4L * 512, 128, 4L * 512, 128,
          1024, 4, 0, 1.f, 0);
    gemmP(st, tmpA, p.wo, p.bo, xc, xo, 4096, 512, 512, 1.f, 0);
    { float* s = xc; xc = xo; xo = s; }
  }

  // ---- z = LN(feat + x1); AE encoder/decoder
  k_add_ln<<<4096, 128, 0, st>>>(xc, featB, ae_lng, ae_lnb, tmpA, 512);
  gemmP(st, tmpA, ew0, eb0, nullptr, bufQ, 4096, 256, 512, 1.f, 1);
  gemmP(st, bufQ, ew1, eb1, nullptr, bufK, 4096, 64, 256, 1.f, 1);
  gemmP(st, bufK, ew2, eb2, nullptr, bufV, 4096, 1, 64, 1.f, 3);
  gemmP(st, bufV, dw0, db0, nullptr, bufQ, 4096, 64, 1, 1.f, 1);
  gemmP(st, bufQ, dw1, db1, nullptr, bufK, 4096, 256, 64, 1.f, 1);
  gemmP(st, bufK, dw2, db2, nullptr, tmpA, 4096, 512, 256, 1.f, 1);
  gemmP(st, tmpA, dw3, db3, nullptr, xA, 4096, 512, 512, 1.f, 1);
  k_add_pe512<<<(4096 * 512 + 255) / 256, 256, 0, st>>>(xA, xB, 4096 * 512);

  // ---- te2 encoder layer (h=8, dk=64)
  gemmP(st, xB, te.wq, te.bq, nullptr, bufQ, 4096, 512, 512, 1.f, 0);
  gemmP(st, xB, te.wk, te.bk, nullptr, bufK, 4096, 512, 512, 1.f, 0);
  gemmP(st, xB, te.wv, te.bv, nullptr, bufV, 4096, 512, 512, 1.f, 0);
  gemmL(st, bufQ, bufK, nullptr, nullptr, scoreB, 128, 128, 64,
        512, 512, 128, 128L * 512, 64, 128L * 512, 64, 8L * 128 * 128, 128L * 128,
        32, 8, 1, 0.125f, 0);
  k_softmax<<<32 * 8 * 128, 128, 0, st>>>(scoreB, 128);
  gemmL(st, scoreB, bufV, nullptr, nullptr, tmpA, 128, 64, 128,
        128, 512, 512, 8L * 128 * 128, 128L * 128, 128L * 512, 64, 128L * 512, 64,
        32, 8, 0, 1.f, 0);
  gemmP(st, tmpA, te.wo, te.bo, xB, xA, 4096, 512, 512, 1.f, 0);
  k_add_ln<<<4096, 128, 0, st>>>(xA, nullptr, te.l1g, te.l1b, xB, 512);
  gemmP(st, xB, te.w1, te.b1, nullptr, bufQ, 4096, 512, 512, 1.f, 1);
  gemmP(st, bufQ, te.w2, te.b2, xB, xA, 4096, 512, 512, 1.f, 0);
  k_add_ln<<<4096, 128, 0, st>>>(xA, nullptr, te.l2g, te.l2b, (float*)d_out, 512);
}